// BoundaryLoss_28071906246697
// MI455X (gfx1250) — compile-verified
//
#include <hip/hip_runtime.h>
#include <math.h>
#include <stdint.h>

#define NN 8
#define CC 4
#define HH 256
#define WW 256

typedef __attribute__((ext_vector_type(2))) float v2f;
typedef __attribute__((ext_vector_type(8))) float v8f;

// Exact 32-lane sum using V_WMMA_F32_16X16X4_F32.
// A (16x4): lane l puts its partial at row (l%16), K = 0 (lanes 0-15) / K = 2 (lanes 16-31),
// the other K slot is 0.  B (4x16) = all ones.  D[m,n] = partial[m] + partial[m+16].
// Lane l<16 holds D[0..7, l] in its 8 C VGPRs; lane l+16 holds D[8..15, l].
// Sum of own 8 VGPRs + shfl_xor(16) => full 32-lane total in every lane.
__device__ __forceinline__ float wave_reduce32(float partial) {
    v2f a; a[0] = partial; a[1] = 0.0f;
    v2f b; b[0] = 1.0f;    b[1] = 1.0f;
    v8f c = {0.f, 0.f, 0.f, 0.f, 0.f, 0.f, 0.f, 0.f};
    c = __builtin_amdgcn_wmma_f32_16x16x4_f32(false, a, false, b, (short)0, c, false, false);
    float s = ((c[0] + c[1]) + (c[2] + c[3])) + ((c[4] + c[5]) + (c[6] + c[7]));
    s += __shfl_xor(s, 16, 32);
    return s;
}

// Kernel 1: per-row 1D nearest-zero distances for all 4 classes, both polarities.
// pos[c] = dist to nearest pixel with target != c (zeros of mask)
// neg[c] = dist to nearest pixel with target == c (zeros of complement mask)
// Both capped at 512 (= H+W, matching the reference BIG).  Packed as u16 pair.
__global__ void row_dist_kernel(const int* __restrict__ target,
                                uint32_t* __restrict__ gpk) {
    const int j  = threadIdx.x;      // 0..255
    const int nh = blockIdx.x;       // n*H + h
    const int n  = nh >> 8;
    const int h  = nh & 255;

    __shared__ int trow[WW];
    trow[j] = target[nh * WW + j];
    __syncthreads();

    int pos[CC], neg[CC];
#pragma unroll
    for (int c = 0; c < CC; ++c) { pos[c] = 512; neg[c] = 512; }

    for (int p = 0; p < WW; ++p) {
        int tp = trow[p];
        int ad = j - p; ad = (ad < 0) ? -ad : ad;
#pragma unroll
        for (int c = 0; c < CC; ++c) {
            bool eq  = (tp == c);
            int mnp  = (ad < pos[c]) ? ad : pos[c];
            int mnn  = (ad < neg[c]) ? ad : neg[c];
            pos[c] = eq ? pos[c] : mnp;   // update pos where t != c
            neg[c] = eq ? mnn    : neg[c]; // update neg where t == c
        }
    }

#pragma unroll
    for (int c = 0; c < CC; ++c) {
        uint32_t v = (uint32_t)pos[c] | ((uint32_t)neg[c] << 16);
        gpk[(((size_t)(c * NN + n) * HH) + h) * WW + j] = v;
    }
}

// Kernel 2: column pass of the EDT (min-plus over k), fused with softmax and
// the probs*sdt dot product.  One block per (mask = (c,n), 16-column strip).
__global__ void col_pass_kernel(const float* __restrict__ input,
                                const uint32_t* __restrict__ gpk,
                                float* __restrict__ partials) {
    const int tid   = threadIdx.x;          // 0..255
    const int jl    = tid & 15;
    const int ibase = (tid >> 4) * 16;      // 16 consecutive rows per thread
    const int j0    = blockIdx.x * 16;      // column strip
    const int m     = blockIdx.y;           // 0..31
    const int c     = m >> 3;               // m / NN
    const int n     = m & 7;                // m % NN

    __shared__ uint32_t gtile[HH * 16];     // 16 KB: g[k][jl] packed pos/neg

    const uint32_t* gbase = gpk + (size_t)(c * NN + n) * HH * WW;
#pragma unroll
    for (int r = 0; r < 16; ++r) {
        int idx = tid + 256 * r;            // 0..4095
        int k   = idx >> 4;
        int jj  = idx & 15;
        gtile[idx] = gbase[(size_t)k * WW + j0 + jj];
    }
    __syncthreads();

    float dp[16], dn[16];
#pragma unroll
    for (int ii = 0; ii < 16; ++ii) { dp[ii] = 1e30f; dn[ii] = 1e30f; }

    for (int k = 0; k < HH; ++k) {
        uint32_t g = gtile[k * 16 + jl];
        float gp  = (float)(g & 0xffffu);
        float gn  = (float)(g >> 16);
        float gp2 = gp * gp;
        float gn2 = gn * gn;
#pragma unroll
        for (int ii = 0; ii < 16; ++ii) {
            float d  = (float)(ibase + ii - k);
            float d2 = d * d;
            dp[ii] = fminf(dp[ii], gp2 + d2);
            dn[ii] = fminf(dn[ii], gn2 + d2);
        }
    }

    const int j = j0 + jl;
    float acc = 0.0f;
#pragma unroll
    for (int ii = 0; ii < 16; ++ii) {
        int i = ibase + ii;
        size_t base = ((size_t)n * CC * HH + i) * WW + j;
        float x0 = input[base + (size_t)0 * HH * WW];
        float x1 = input[base + (size_t)1 * HH * WW];
        float x2 = input[base + (size_t)2 * HH * WW];
        float x3 = input[base + (size_t)3 * HH * WW];
        float mx = fmaxf(fmaxf(x0, x1), fmaxf(x2, x3));
        float e0 = expf(x0 - mx), e1 = expf(x1 - mx);
        float e2 = expf(x2 - mx), e3 = expf(x3 - mx);
        float den = (e0 + e1) + (e2 + e3);
        float ec  = (c == 0) ? e0 : (c == 1) ? e1 : (c == 2) ? e2 : e3;
        float pc  = ec / den;
        float sdt = sqrtf(dp[ii]) - sqrtf(dn[ii]);
        acc += pc * sdt;
    }

    // Block reduction: WMMA per wave, then 8 wave sums.
    float wsum = wave_reduce32(acc);
    __shared__ float warr[8];
    int lane = tid & 31, wid = tid >> 5;
    if (lane == 0) warr[wid] = wsum;
    __syncthreads();
    if (tid == 0) {
        float b = 0.0f;
#pragma unroll
        for (int w = 0; w < 8; ++w) b += warr[w];
        partials[blockIdx.y * 16 + blockIdx.x] = b;
    }
}

// Kernel 3: reduce 512 block partials to the scalar loss.
__global__ void final_reduce_kernel(const float* __restrict__ partials,
                                    float* __restrict__ out) {
    const int tid = threadIdx.x;            // 0..255
    float s = partials[tid] + partials[tid + 256];
    float wsum = wave_reduce32(s);
    __shared__ float warr[8];
    int lane = tid & 31, wid = tid >> 5;
    if (lane == 0) warr[wid] = wsum;
    __syncthreads();
    if (tid == 0) {
        float t = 0.0f;
#pragma unroll
        for (int w = 0; w < 8; ++w) t += warr[w];
        double r = (double)t / 32.0 / (65536.0 + 1e-6);
        out[0] = (float)r;
    }
}

extern "C" void kernel_launch(void* const* d_in, const int* in_sizes, int n_in,
                              void* d_out, int out_size, void* d_ws, size_t ws_size,
                              hipStream_t stream) {
    (void)in_sizes; (void)n_in; (void)out_size; (void)ws_size;
    const float* input  = (const float*)d_in[0];  // [N,C,H,W] fp32
    const int*   target = (const int*)d_in[1];    // [N,H,W]   int32

    uint32_t* gpk = (uint32_t*)d_ws;  // [C*N, H, W] packed u16 pos|neg : 8 MB
    float* partials = (float*)((char*)d_ws +
                               (size_t)CC * NN * HH * WW * sizeof(uint32_t)); // 512 floats

    row_dist_kernel<<<NN * HH, WW, 0, stream>>>(target, gpk);

    dim3 g2(WW / 16, CC * NN);   // 16 strips x 32 masks = 512 blocks
    col_pass_kernel<<<g2, 256, 0, stream>>>(input, gpk, partials);

    final_reduce_kernel<<<1, 256, 0, stream>>>(partials, (float*)d_out);
}